// SpatialAttention_1580547971020
// MI455X (gfx1250) — compile-verified
//
#include <hip/hip_runtime.h>

typedef __attribute__((ext_vector_type(2))) float v2f;
typedef __attribute__((ext_vector_type(4))) float v4f;
typedef __attribute__((ext_vector_type(8))) float v8f;

#define BATCH 8
#define NTOK  2048
#define EDIM  64
#define DDIM  128

#define ESTRIDE 68    // 64 + 4 pad: conflict-free stage-1 B reads, 16B-aligned rows
#define ZROWS   136   // 128 + 8 pad: 2*136 mod 64 == 16 -> half-waves hit disjoint banks
#define PSTRIDE 18    // 16 + 2 pad: aligned float2 A-layout reads of P

// ---------------------------------------------------------------------------
// Kernel 1: per-row squared L2 norms, sq[b*N + i] = ||e_i||^2
// ---------------------------------------------------------------------------
__global__ __launch_bounds__(256) void sq_kernel(const float* __restrict__ e,
                                                 float* __restrict__ sq) {
    int idx = blockIdx.x * 256 + threadIdx.x;
    const float* row = e + (size_t)idx * EDIM;
    float acc = 0.f;
#pragma unroll
    for (int q = 0; q < EDIM / 4; ++q) {
        v4f v = *(const v4f*)(row + 4 * q);
        acc += v.x * v.x + v.y * v.y + v.z * v.z + v.w * v.w;
    }
    sq[idx] = acc;
}

// 16-byte async copy global -> LDS (CDNA5 GLOBAL_LOAD_ASYNC_TO_LDS_B128)
typedef int v4i_g __attribute__((vector_size(16)));
__device__ __forceinline__ void async_copy16(const float* g, float* l) {
    __builtin_amdgcn_global_load_async_to_lds_b128(
        (__attribute__((address_space(1))) v4i_g*)g,
        (__attribute__((address_space(3))) v4i_g*)l,
        /*offset=*/0, /*cpol=*/0);
}

// ---------------------------------------------------------------------------
// Kernel 2: fused spatial attention, flash-style over j tiles.
// Double-buffered LDS fed by async-to-LDS copies; one barrier per iteration.
// ---------------------------------------------------------------------------
__global__ __launch_bounds__(128) void attn_kernel(const float* __restrict__ E,
                                                   const float* __restrict__ Z,
                                                   const float* __restrict__ SQ,
                                                   float* __restrict__ out) {
    __shared__ float ej[2][16 * ESTRIDE];    // E_j tile, row major
    __shared__ float zj[2][16 * ZROWS];      // Z_j tile, row major
    __shared__ float pbuf[4][16 * PSTRIDE];  // per-wave P tile [i][j]

    const int b    = blockIdx.y;
    const int iblk = blockIdx.x;              // 0..31
    const int t    = threadIdx.x;             // 0..127
    const int wave = t >> 5;
    const int lane = t & 31;
    const int n    = lane & 15;
    const int hi   = lane >> 4;

    const int i0 = iblk * 64 + wave * 16;

    const float* Eb  = E  + (size_t)b * NTOK * EDIM;
    const float* Zb  = Z  + (size_t)b * NTOK * DDIM;
    const float* SQb = SQ + (size_t)b * NTOK;

    // Per-thread async-copy slots (2 for E tile, 4 for Z tile)
    const int ei0 = t * 2;                    // 0..255 -> jr = ei>>4, col = (ei&15)*4
    const int zi0 = t * 4;                    // 0..511 -> jr = zi>>5, col = (zi&31)*4

    // ---- A tile: rows i0..i0+15 of E in WMMA f32 16x16x4 A-layout ----------
    v2f areg[16];
#pragma unroll
    for (int kc = 0; kc < 16; ++kc) {
        const int kb = kc * 4 + hi * 2;
        areg[kc] = *(const v2f*)(Eb + (size_t)(i0 + n) * EDIM + kb);
    }
    float sqi[8];
#pragma unroll
    for (int r = 0; r < 8; ++r) sqi[r] = SQb[i0 + r + hi * 8];

    v8f   oacc[8] = {};
    float rsum[8] = {};

    // ---- pipeline prologue: async-stage tile 0 into buffer 0 ----
#pragma unroll
    for (int q = 0; q < 2; ++q) {
        const int idx = ei0 + q, jr = idx >> 4, c0 = (idx & 15) * 4;
        async_copy16(Eb + (size_t)jr * EDIM + c0, &ej[0][jr * ESTRIDE + c0]);
    }
#pragma unroll
    for (int q = 0; q < 4; ++q) {
        const int idx = zi0 + q, jr = idx >> 5, c0 = (idx & 31) * 4;
        async_copy16(Zb + (size_t)jr * DDIM + c0, &zj[0][jr * ZROWS + c0]);
    }
    float sqj_cur = SQb[n];
    __builtin_amdgcn_s_wait_asynccnt(0);
    __syncthreads();

    for (int jt = 0; jt < NTOK / 16; ++jt) {
        const int buf = jt & 1;
        const float* ejb = ej[buf];
        const float* zjb = zj[buf];

        // ---- issue next tile's async copies into the free buffer ----------
        const bool have_next = (jt + 1) < (NTOK / 16);
        float sqj_nxt = sqj_cur;
        if (have_next) {
            const int j0n = (jt + 1) * 16;
            float* ejn = ej[buf ^ 1];
            float* zjn = zj[buf ^ 1];
#pragma unroll
            for (int q = 0; q < 2; ++q) {
                const int idx = ei0 + q, jr = idx >> 4, c0 = (idx & 15) * 4;
                async_copy16(Eb + (size_t)(j0n + jr) * EDIM + c0,
                             &ejn[jr * ESTRIDE + c0]);
            }
#pragma unroll
            for (int q = 0; q < 4; ++q) {
                const int idx = zi0 + q, jr = idx >> 5, c0 = (idx & 31) * 4;
                async_copy16(Zb + (size_t)(j0n + jr) * DDIM + c0,
                             &zjn[jr * ZROWS + c0]);
            }
            sqj_nxt = SQb[j0n + n];
        }

        // ---- stage 1: S = E_i . E_j^T, two independent WMMA chains --------
        v8f s0 = {}, s1 = {};
#pragma unroll
        for (int kc = 0; kc < 16; kc += 2) {
            v2f b0 = *(const v2f*)&ejb[n * ESTRIDE + (kc + 0) * 4 + hi * 2];
            v2f b1 = *(const v2f*)&ejb[n * ESTRIDE + (kc + 1) * 4 + hi * 2];
            s0 = __builtin_amdgcn_wmma_f32_16x16x4_f32(
                     false, areg[kc + 0], false, b0, (short)0, s0, false, false);
            s1 = __builtin_amdgcn_wmma_f32_16x16x4_f32(
                     false, areg[kc + 1], false, b1, (short)0, s1, false, false);
        }

        // ---- similarity: p = exp(-max(||i||^2 + ||j||^2 - 2*inner, 0)) ----
        float pv[8];
#pragma unroll
        for (int r = 0; r < 8; ++r) {
            float inner = s0[r] + s1[r];
            float d2 = fmaxf(sqi[r] + sqj_cur - 2.0f * inner, 0.0f);
            float p  = __expf(-d2);
            pv[r]    = p;
            rsum[r] += p;
        }
#pragma unroll
        for (int r = 0; r < 8; ++r)
            pbuf[wave][(r + hi * 8) * PSTRIDE + n] = pv[r];
        // per-wave LDS ops are in order; no workgroup barrier needed here

        // ---- stage 2: O += P . Z_j --------------------------------------
#pragma unroll
        for (int kc = 0; kc < 4; ++kc) {
            const int k = kc * 4 + hi * 2;
            v2f pa = *(const v2f*)&pbuf[wave][n * PSTRIDE + k];
            const float* zrow = &zjb[k * ZROWS + n];   // one addr VGPR per kc
            v2f zb[8];
#pragma unroll
            for (int dc = 0; dc < 8; ++dc) {
                zb[dc].x = zrow[dc * 16];              // row k   (imm offset)
                zb[dc].y = zrow[dc * 16 + ZROWS];      // row k+1 (imm offset)
            }
#pragma unroll
            for (int dc = 0; dc < 8; ++dc)
                oacc[dc] = __builtin_amdgcn_wmma_f32_16x16x4_f32(
                               false, pa, false, zb[dc], (short)0, oacc[dc],
                               false, false);
        }

        // ---- close pipeline stage: my asyncs done, then WG-wide visible ---
        __builtin_amdgcn_s_wait_asynccnt(0);
        __syncthreads();
        sqj_cur = sqj_nxt;
    }

    // ---- reduce row sums across each 16-lane half (wave32 xor shuffles) ----
#pragma unroll
    for (int r = 0; r < 8; ++r) {
        float v = rsum[r];
#pragma unroll
        for (int off = 8; off >= 1; off >>= 1)
            v += __shfl_xor(v, off, 32);
        rsum[r] = v;   // row total for M = r + hi*8
    }

    // ---- normalize and store ----
    float* ob = out + (size_t)b * NTOK * DDIM;
#pragma unroll
    for (int r = 0; r < 8; ++r) {
        const float inv = 1.0f / rsum[r];
        const int   row = i0 + r + hi * 8;
#pragma unroll
        for (int dc = 0; dc < 8; ++dc)
            ob[(size_t)row * DDIM + dc * 16 + n] = oacc[dc][r] * inv;
    }
}

// ---------------------------------------------------------------------------
extern "C" void kernel_launch(void* const* d_in, const int* in_sizes, int n_in,
                              void* d_out, int out_size, void* d_ws, size_t ws_size,
                              hipStream_t stream) {
    const float* E = (const float*)d_in[0];   // [B, N, 64]
    const float* Z = (const float*)d_in[1];   // [B, N, 128]
    float* out = (float*)d_out;               // [B, N, 128]
    float* sq  = (float*)d_ws;                // [B*N] scratch row norms

    hipLaunchKernelGGL(sq_kernel, dim3(BATCH * NTOK / 256), dim3(256), 0, stream,
                       E, sq);
    hipLaunchKernelGGL(attn_kernel, dim3(NTOK / 64, BATCH), dim3(128), 0, stream,
                       E, Z, sq, out);
}